// Attention_Inner_Product_61873298866223
// MI455X (gfx1250) — compile-verified
//
#include <hip/hip_runtime.h>
#include <hip/hip_bf16.h>

// Problem constants (match reference)
#define N_NODES 50000
#define N_EDGES 800000
#define FEAT    64
#define HEADS   4

typedef float v2f __attribute__((ext_vector_type(2)));
typedef float v8f __attribute__((ext_vector_type(8)));

// Workspace layout (in floats):
//   [0]                : encoded global-max (as unsigned)
//   [16 .. )           : denom [N_NODES*HEADS]
//   [16+N*H .. )       : Y     [N_NODES*HEADS*FEAT]  (node transform)
#define WS_DENOM 16
#define WS_Y     (16 + N_NODES * HEADS)

// Order-preserving float<->uint mapping so atomicMax(uint) == float max,
// and memset-0 is an identity element (0 < enc(-inf)).
__device__ __forceinline__ unsigned enc_f32(float f) {
  unsigned u = __float_as_uint(f);
  return (u & 0x80000000u) ? ~u : (u | 0x80000000u);
}
__device__ __forceinline__ float dec_f32(unsigned u) {
  unsigned b = (u & 0x80000000u) ? (u ^ 0x80000000u) : ~u;
  return __uint_as_float(b);
}

// ---------------------------------------------------------------------------
// Kernel 1: Y[n,h,f] = sum_g W[h,f,g] * X[n,g]  via V_WMMA_F32_16X16X4_F32.
// One wave32 handles one 16-node tile. X tile staged in LDS (pad stride 68).
// A-frag (16x4 f32): lanes 0-15 rows M, VGPR pair holds K; half-wave = K+2.
// B-frag (4x16 f32): B[g,f] = W[h][f][g], lane = column f, K split likewise.
// C/D (16x16 f32): lane = N col, VGPR j = row M (second half-wave: M+8).
// ---------------------------------------------------------------------------
#define XPAD 68
__global__ __launch_bounds__(128) void node_transform_wmma(
    const float* __restrict__ x, const float* __restrict__ w,
    float* __restrict__ y) {
  __shared__ float lx[4][16 * XPAD];
  const int wave = threadIdx.x >> 5;
  const int lane = threadIdx.x & 31;
  const int tile = blockIdx.x * 4 + wave;
  const bool active = (tile * 16) < N_NODES;   // wave-uniform -> EXEC all-1s

  if (active) {
    // Coalesced stage of 16x64 X tile (1024 contiguous floats).
    const float* srcp = x + (size_t)tile * 16 * FEAT;
#pragma unroll
    for (int i = 0; i < 8; ++i) {
      const int fidx = i * 128 + lane * 4;
      const int r = fidx >> 6, c = fidx & 63;
      *(float4*)(&lx[wave][r * XPAD + c]) = *(const float4*)(srcp + fidx);
    }
  }
  __syncthreads();
  if (!active) return;

  const int row   = lane & 15;
  const int khalf = (lane >> 4) << 1;  // 0 or 2 (half-wave K split)

  // Hoist all 16 A fragments (K = 0..63 in steps of 4) into registers.
  v2f a[16];
#pragma unroll
  for (int k4 = 0; k4 < 16; ++k4)
    a[k4] = *(const v2f*)(&lx[wave][row * XPAD + k4 * 4 + khalf]);

  const int nbase = tile * 16;
#pragma unroll
  for (int h = 0; h < HEADS; ++h) {
#pragma unroll
    for (int nt = 0; nt < 4; ++nt) {
      const int f0   = nt * 16;
      const int fcol = f0 + (lane & 15);
      // B[g, fcol] = W[h][fcol][g]; W cached in L2 (64 KB total).
      const float* wb = w + ((size_t)h * FEAT + fcol) * FEAT + khalf;
      v8f c = {0.f, 0.f, 0.f, 0.f, 0.f, 0.f, 0.f, 0.f};
#pragma unroll
      for (int k4 = 0; k4 < 16; ++k4) {
        const v2f b = *(const v2f*)(wb + k4 * 4);
        c = __builtin_amdgcn_wmma_f32_16x16x4_f32(
            /*neg_a=*/false, a[k4], /*neg_b=*/false, b,
            /*c_mod=*/(short)0, c, /*reuse_a=*/false, /*reuse_b=*/false);
      }
      const int ncol = lane & 15;
      const int moff = (lane >> 4) * 8;
#pragma unroll
      for (int j = 0; j < 8; ++j) {
        const int node = nbase + moff + j;
        y[(size_t)node * (HEADS * FEAT) + h * FEAT + f0 + ncol] = c[j];
      }
    }
  }
}

// ---------------------------------------------------------------------------
// Kernel 2: per-edge scores. 16 lanes per edge (16 edges per 256-thr block).
// vals[e,h] = <x[src], Y[trg,h,:]>; LeakyReLU(0.2); raw scores -> attention
// output region (scratch); x_src -> second output region; global encoded max.
// ---------------------------------------------------------------------------
__global__ __launch_bounds__(256) void edge_scores(
    const float* __restrict__ x, const int* __restrict__ eidx,
    const float* __restrict__ y, float* __restrict__ out_scores,
    float* __restrict__ out_xsrc, unsigned* __restrict__ gmax) {
  __shared__ unsigned smax;
  if (threadIdx.x == 0) smax = 0u;
  __syncthreads();

  const int grp = threadIdx.x >> 4;
  const int l   = threadIdx.x & 15;
  const int e   = blockIdx.x * 16 + grp;  // E = 50000*16 exactly

  const int trg = eidx[e];
  const int src = eidx[N_EDGES + e];

  const float4 xs = *(const float4*)(x + (size_t)src * FEAT + l * 4);
  *(float4*)(out_xsrc + (size_t)e * FEAT + l * 4) = xs;   // x_src output

  float p[HEADS];
#pragma unroll
  for (int h = 0; h < HEADS; ++h) {
    const float4 ys =
        *(const float4*)(y + (size_t)trg * (HEADS * FEAT) + h * FEAT + l * 4);
    p[h] = xs.x * ys.x + xs.y * ys.y + xs.z * ys.z + xs.w * ys.w;
  }
#pragma unroll
  for (int h = 0; h < HEADS; ++h) {
#pragma unroll
    for (int off = 8; off >= 1; off >>= 1)
      p[h] += __shfl_xor(p[h], off, 16);
    p[h] = (p[h] > 0.f) ? p[h] : 0.2f * p[h];  // LeakyReLU(0.2)
  }
  if (l < HEADS) out_scores[e * HEADS + l] = p[l];

  const float m = fmaxf(fmaxf(p[0], p[1]), fmaxf(p[2], p[3]));
  atomicMax(&smax, enc_f32(m));
  __syncthreads();
  if (threadIdx.x == 0) atomicMax(gmax, smax);
}

// ---------------------------------------------------------------------------
// Kernel 3: exp(score - gmax) in place; accumulate denom[trg,h] via atomics.
// ---------------------------------------------------------------------------
__global__ __launch_bounds__(256) void edge_exp_denom(
    const int* __restrict__ eidx, const unsigned* __restrict__ gmax_enc,
    float* __restrict__ scores_inout, float* __restrict__ denom) {
  const int tid = blockIdx.x * 256 + threadIdx.x;  // == E*H exactly
  const float gmax = dec_f32(*gmax_enc);
  const int e = tid >> 2;
  const int h = tid & 3;
  const float ex = __expf(scores_inout[tid] - gmax);
  scores_inout[tid] = ex;
  atomicAdd(&denom[(size_t)eidx[e] * HEADS + h], ex);
}

// ---------------------------------------------------------------------------
// Kernel 4: normalize in place: att = exp / (denom[trg,h] + 1e-16).
// ---------------------------------------------------------------------------
__global__ __launch_bounds__(256) void edge_norm(
    const int* __restrict__ eidx, const float* __restrict__ denom,
    float* __restrict__ att) {
  const int tid = blockIdx.x * 256 + threadIdx.x;
  const int e = tid >> 2;
  const int h = tid & 3;
  att[tid] = att[tid] / (denom[(size_t)eidx[e] * HEADS + h] + 1e-16f);
}

// ---------------------------------------------------------------------------
extern "C" void kernel_launch(void* const* d_in, const int* in_sizes, int n_in,
                              void* d_out, int out_size, void* d_ws,
                              size_t ws_size, hipStream_t stream) {
  const float* x   = (const float*)d_in[0];  // nodes_features [N,1,F]
  const int* eidx  = (const int*)d_in[1];    // edge_index [2,E]
  const float* w   = (const float*)d_in[2];  // metric_weights [1,H,F,F]

  float* out = (float*)d_out;
  float* ws  = (float*)d_ws;

  unsigned* gmax = (unsigned*)ws;            // ws[0]
  float* denom   = ws + WS_DENOM;            // [N*H]
  float* y       = ws + WS_Y;                // [N*H*F]
  float* att     = out;                      // [E*H]  (also score scratch)
  float* xsrc    = out + (size_t)N_EDGES * HEADS;  // [E*F]

  // Zero max slot + denominators (graph-capturable memset node).
  hipMemsetAsync(d_ws, 0, (size_t)(WS_Y) * sizeof(float), stream);

  node_transform_wmma<<<(N_NODES / 16 + 3) / 4, 128, 0, stream>>>(x, w, y);
  edge_scores<<<N_EDGES / 16, 256, 0, stream>>>(x, eidx, y, att, xsrc, gmax);
  edge_exp_denom<<<(N_EDGES * HEADS) / 256, 256, 0, stream>>>(eidx, gmax, att,
                                                              denom);
  edge_norm<<<(N_EDGES * HEADS) / 256, 256, 0, stream>>>(eidx, denom, att);
}